// NormActivation_90838558311180
// MI455X (gfx1250) — compile-verified
//
#include <hip/hip_runtime.h>
#include <math.h>
#include <stdint.h>

#define DIM      1184      // floats per row = 256 + 384 + 320 + 224
#define NVEC     296       // float4s per row
#define WPB      4         // waves per block
#define BLOCK    (WPB*32)
#define NBLOCKS  2048      // 8192 waves -> 8 rows per wave

typedef __attribute__((ext_vector_type(4))) float v4f;   // native vector: OK for
                                                         // __builtin_nontemporal_store

// Truncating a generic pointer to __shared__ data yields the 32-bit LDS byte
// address (AMDGPU generic LDS pointer = {shared aperture hi32, lds offset lo32}).
__device__ __forceinline__ unsigned lds_addr_of(const void* p) {
    return (unsigned)(uintptr_t)p;
}

// CDNA5 async DMA: copies 16 bytes from global address vaddr+off to LDS address vdst+off, tracked by ASYNCcnt.
// The same immediate offset applies to BOTH global and LDS addresses (ISA 08 §4.4),
// so one base address pair covers a whole row. th:TH_LOAD_NT: stream-once data,
// don't cache in WGP$/L2 (dataset is 620 MB vs 192 MB L2, zero reuse).
template<int OFF>
__device__ __forceinline__ void async_ld_b128_nt(unsigned lds, const void* g) {
    asm volatile("global_load_async_to_lds_b128 %0, %1, off offset:%2 th:TH_LOAD_NT"
                 :: "v"(lds), "v"(g), "n"(OFF) : "memory");
}

template<int N>
__device__ __forceinline__ void wait_async() {
    asm volatile("s_wait_asynccnt %0" :: "n"(N) : "memory");
}

__device__ __forceinline__ void wait_ds0() {
    asm volatile("s_wait_dscnt 0" ::: "memory");
}

__device__ __forceinline__ float wave_sum32(float s) {
#pragma unroll
    for (int m = 16; m > 0; m >>= 1) s += __shfl_xor(s, m, 32);
    return s;
}

extern "C" __global__ __launch_bounds__(BLOCK, 1)
void norm_act_kernel(const float* __restrict__ x, float* __restrict__ y, int nrows)
{
    __shared__ v4f lds_buf[WPB * 2 * NVEC];   // 37888 bytes

    const int lane = threadIdx.x & 31;
    const int wv   = threadIdx.x >> 5;
    const int gw   = blockIdx.x * WPB + wv;
    const int nw   = gridDim.x * WPB;

    v4f* buf0 = &lds_buf[(wv * 2 + 0) * NVEC];
    v4f* buf1 = &lds_buf[(wv * 2 + 1) * NVEC];
    const unsigned l0 = lds_addr_of(buf0 + lane);
    const unsigned l1 = lds_addr_of(buf1 + lane);

    int row = gw;
    if (row >= nrows) return;

    // Issue one row (10 async b128 loads) into buffer b. Count is uniform:
    // the tail instruction executes with EXEC=0xFF (lanes 0..7), still +1 ASYNCcnt.
    auto issue = [&](int r, int b) {
        wait_ds0();  // ensure prior ds reads of this buffer retired (WAR safety)
        const v4f* g = (const v4f*)(x + (size_t)r * DIM) + lane;
        unsigned l = b ? l1 : l0;
        async_ld_b128_nt<0   >(l, g);
        async_ld_b128_nt<512 >(l, g);
        async_ld_b128_nt<1024>(l, g);
        async_ld_b128_nt<1536>(l, g);
        async_ld_b128_nt<2048>(l, g);
        async_ld_b128_nt<2560>(l, g);
        async_ld_b128_nt<3072>(l, g);
        async_ld_b128_nt<3584>(l, g);
        async_ld_b128_nt<4096>(l, g);
        if (lane < 8) async_ld_b128_nt<4608>(l, g);  // tail: vecs 288..295
    };

    issue(row, 0);
    int cur = 0;

    while (true) {
        const int nxt = row + nw;
        const bool have = nxt < nrows;
        if (have) { issue(nxt, cur ^ 1); wait_async<10>(); }  // 10 newest pending
        else      { wait_async<0>(); }

        const v4f* B = cur ? buf1 : buf0;

        v4f v[10];
        float s0 = 0.f, s1 = 0.f, s2 = 0.f, s3 = 0.f;
#pragma unroll
        for (int k = 0; k < 9; ++k) {
            const int vi = k * 32 + lane;
            v4f t = B[vi];
            v[k] = t;
            const float ss = t.x*t.x + t.y*t.y + t.z*t.z + t.w*t.w;
            if      (vi <  64) s0 += ss;   // constant-folds for all k except k==7
            else if (vi < 160) s1 += ss;
            else if (vi < 240) s2 += ss;
            else               s3 += ss;
        }
        if (lane < 8) {
            v4f t = B[288 + lane];
            v[9] = t;
            s3 += t.x*t.x + t.y*t.y + t.z*t.z + t.w*t.w;
        }

        s0 = wave_sum32(s0); s1 = wave_sum32(s1);
        s2 = wave_sum32(s2); s3 = wave_sum32(s3);

        const float n0 = s0 > 0.f ? sqrtf(s0 * (1.f / 256.f)) : 1.f;
        const float n1 = s1 > 0.f ? sqrtf(s1 * (1.f / 384.f)) : 1.f;
        const float n2 = s2 > 0.f ? sqrtf(s2 * (1.f / 320.f)) : 1.f;
        const float n3 = s3 > 0.f ? sqrtf(s3 * (1.f / 224.f)) : 1.f;

        const float f0 = n0 / (1.f + __expf(-n0));        // silu
        const float f1 = 1.f / (1.f + __expf(-n1));       // sigmoid
        const float e2 = __expf(-2.f * n2);
        const float f2 = (1.f - e2) / (1.f + e2);         // tanh (n2 >= 0)
        const float f3 = n3 / (1.f + __expf(-n3));        // silu

        v4f* dst = (v4f*)(y + (size_t)row * DIM);
#pragma unroll
        for (int k = 0; k < 9; ++k) {
            const int vi = k * 32 + lane;
            const float f = (vi < 64) ? f0 : (vi < 160) ? f1 : (vi < 240) ? f2 : f3;
            v4f t = v[k] * f;
            __builtin_nontemporal_store(t, &dst[vi]);     // store th:TH_NT
        }
        if (lane < 8) {
            v4f t = v[9] * f3;
            __builtin_nontemporal_store(t, &dst[288 + lane]);
        }

        if (!have) break;
        row = nxt;
        cur ^= 1;
    }
}

extern "C" void kernel_launch(void* const* d_in, const int* in_sizes, int n_in,
                              void* d_out, int out_size, void* d_ws, size_t ws_size,
                              hipStream_t stream) {
    const float* x = (const float*)d_in[0];
    float*       y = (float*)d_out;
    const int nrows = in_sizes[0] / DIM;   // 65536

    int nblocks = NBLOCKS;
    if (nblocks * WPB > nrows) nblocks = (nrows + WPB - 1) / WPB;

    norm_act_kernel<<<nblocks, BLOCK, 0, stream>>>(x, y, nrows);
}